// RouterLoss_53532472377600
// MI455X (gfx1250) — compile-verified
//
#include <hip/hip_runtime.h>

// Router loss: single streaming pass (HBM-bound, 256MB / 23.3 TB/s ~= 11us floor).
// - WMMA f32 16x16x4 (A = ones) as the reduction engine for sum(p) and sum((p-l)^2)
// - LDS ds_add_f32/ds_add_u32 atomics for the 64-seed segment sums
// - non-temporal b128 loads (working set > 192MB L2, read-once)
// - uniform (scalar) trip count in the main loop so EXEC stays all-ones for WMMA
//   (ISA 7.12 restriction) and the loop branch is pure SALU.

#define GRID        1024
#define BLOCK       256
#define NSEEDS      64
#define CVAR_K      12          // int(64 * 0.2)
#define PART_STRIDE 132         // per-block floats in ws: [0]=sum_p [1]=sum_brier [2..65]=fail [66..129]=cnt

typedef float f32x4 __attribute__((ext_vector_type(4)));
typedef int   i32x4 __attribute__((ext_vector_type(4)));
typedef float v2f   __attribute__((ext_vector_type(2)));
typedef float v8f   __attribute__((ext_vector_type(8)));

__global__ __launch_bounds__(BLOCK) void rl_partials(
    const float* __restrict__ p, const float* __restrict__ lab,
    const float* __restrict__ suc, const int* __restrict__ seed,
    float* __restrict__ ws, int nvec)
{
    __shared__ float    s_fail[NSEEDS];
    __shared__ unsigned s_cnt[NSEEDS];
    __shared__ float    s_red[BLOCK];

    const int tid = threadIdx.x;
    if (tid < NSEEDS) { s_fail[tid] = 0.0f; s_cnt[tid] = 0u; }
    __syncthreads();

    const v2f ones = {1.0f, 1.0f};
    v8f acc_p = {0.f,0.f,0.f,0.f,0.f,0.f,0.f,0.f};   // WMMA accumulator for sum(p)
    v8f acc_b = {0.f,0.f,0.f,0.f,0.f,0.f,0.f,0.f};   // WMMA accumulator for sum((p-l)^2)
    float lp_tail = 0.0f, lb_tail = 0.0f;            // non-WMMA tail accumulators

    const int idx    = blockIdx.x * BLOCK + tid;
    const int stride = GRID * BLOCK;
    const int full   = nvec / stride;                // scalar-uniform trip count

    for (int it = 0; it < full; ++it) {              // SALU loop, EXEC untouched
        const int i = idx + it * stride;

        // read-once streaming: non-temporal b128 loads
        f32x4 pv = __builtin_nontemporal_load((const f32x4*)p    + i);
        f32x4 lv = __builtin_nontemporal_load((const f32x4*)lab  + i);
        f32x4 sv = __builtin_nontemporal_load((const f32x4*)suc  + i);
        i32x4 dv = __builtin_nontemporal_load((const i32x4*)seed + i);

        // sum(p): ones(16x4) x B(4x16) accumulates column sums into C.
        // Any permutation of the wave's 64 elements into B yields the same total.
        v2f b0 = {pv.x, pv.y};
        v2f b1 = {pv.z, pv.w};
        acc_p = __builtin_amdgcn_wmma_f32_16x16x4_f32(false, ones, false, b0,
                                                      (short)0, acc_p, false, false);
        acc_p = __builtin_amdgcn_wmma_f32_16x16x4_f32(false, ones, false, b1,
                                                      (short)0, acc_p, false, false);

        // brier: sum((p-l)^2)
        f32x4 d = pv - lv;
        v2f q0 = {d.x * d.x, d.y * d.y};
        v2f q1 = {d.z * d.z, d.w * d.w};
        acc_b = __builtin_amdgcn_wmma_f32_16x16x4_f32(false, ones, false, q0,
                                                      (short)0, acc_b, false, false);
        acc_b = __builtin_amdgcn_wmma_f32_16x16x4_f32(false, ones, false, q1,
                                                      (short)0, acc_b, false, false);

        // 64-seed segment sums via LDS atomics (ds_add_f32 / ds_add_u32)
        atomicAdd(&s_fail[dv.x & 63], 1.0f - sv.x); atomicAdd(&s_cnt[dv.x & 63], 1u);
        atomicAdd(&s_fail[dv.y & 63], 1.0f - sv.y); atomicAdd(&s_cnt[dv.y & 63], 1u);
        atomicAdd(&s_fail[dv.z & 63], 1.0f - sv.z); atomicAdd(&s_cnt[dv.z & 63], 1u);
        atomicAdd(&s_fail[dv.w & 63], 1.0f - sv.w); atomicAdd(&s_cnt[dv.w & 63], 1u);
    }

    // tail (never runs for N = 2^24; keeps kernel general without touching WMMA)
    {
        const int i = idx + full * stride;
        if (i < nvec) {
            f32x4 pv = *((const f32x4*)p    + i);
            f32x4 lv = *((const f32x4*)lab  + i);
            f32x4 sv = *((const f32x4*)suc  + i);
            i32x4 dv = *((const i32x4*)seed + i);
            lp_tail += pv.x + pv.y + pv.z + pv.w;
            f32x4 d = pv - lv;
            lb_tail += d.x*d.x + d.y*d.y + d.z*d.z + d.w*d.w;
            atomicAdd(&s_fail[dv.x & 63], 1.0f - sv.x); atomicAdd(&s_cnt[dv.x & 63], 1u);
            atomicAdd(&s_fail[dv.y & 63], 1.0f - sv.y); atomicAdd(&s_cnt[dv.y & 63], 1u);
            atomicAdd(&s_fail[dv.z & 63], 1.0f - sv.z); atomicAdd(&s_cnt[dv.z & 63], 1u);
            atomicAdd(&s_fail[dv.w & 63], 1.0f - sv.w); atomicAdd(&s_cnt[dv.w & 63], 1u);
        }
    }

    // each column-sum is replicated over the 16 rows of C -> sum all, divide by 16
    float lp = 0.0f, lb = 0.0f;
#pragma unroll
    for (int k = 0; k < 8; ++k) { lp += acc_p[k]; lb += acc_b[k]; }
    lp = lp * (1.0f / 16.0f) + lp_tail;
    lb = lb * (1.0f / 16.0f) + lb_tail;

    __syncthreads();                       // bin atomics complete; safe to reuse LDS
    s_red[tid] = lp;
    __syncthreads();
    for (int off = BLOCK / 2; off > 0; off >>= 1) {
        if (tid < off) s_red[tid] += s_red[tid + off];
        __syncthreads();
    }
    if (tid == 0) ws[(size_t)blockIdx.x * PART_STRIDE + 0] = s_red[0];
    __syncthreads();

    s_red[tid] = lb;
    __syncthreads();
    for (int off = BLOCK / 2; off > 0; off >>= 1) {
        if (tid < off) s_red[tid] += s_red[tid + off];
        __syncthreads();
    }
    if (tid == 0) ws[(size_t)blockIdx.x * PART_STRIDE + 1] = s_red[0];

    if (tid < NSEEDS) {
        ws[(size_t)blockIdx.x * PART_STRIDE + 2  + tid] = s_fail[tid];
        ws[(size_t)blockIdx.x * PART_STRIDE + 66 + tid] = (float)s_cnt[tid];
    }
}

__global__ __launch_bounds__(64) void rl_finalize(
    const float* __restrict__ ws, const float* __restrict__ lambda_p,
    float* __restrict__ out, int nblocks, float invN)
{
    __shared__ float s_rate[NSEEDS];
    __shared__ float s_sp[NSEEDS];
    __shared__ float s_sb[NSEEDS];

    const int t = threadIdx.x;

    float f = 0.0f, c = 0.0f;
    for (int b = 0; b < nblocks; ++b) {
        const float* base = ws + (size_t)b * PART_STRIDE;
        f += base[2 + t];
        c += base[66 + t];
    }
    float sp = 0.0f, sb = 0.0f;
    for (int b = t; b < nblocks; b += 64) {
        const float* base = ws + (size_t)b * PART_STRIDE;
        sp += base[0];
        sb += base[1];
    }

    s_rate[t] = f / fmaxf(c, 1.0f);   // counts always > 0 here; guard matches reference
    s_sp[t]   = sp;
    s_sb[t]   = sb;
    __syncthreads();

    for (int off = 32; off > 0; off >>= 1) {
        if (t < off) { s_sp[t] += s_sp[t + off]; s_sb[t] += s_sb[t + off]; }
        __syncthreads();
    }

    if (t == 0) {
        // mean of top-12 of 64 rates (tiny serial selection)
        float acc = 0.0f;
        for (int k = 0; k < CVAR_K; ++k) {
            int   best = 0;
            float bv   = s_rate[0];
            for (int i = 1; i < NSEEDS; ++i) {
                float v = s_rate[i];
                if (v > bv) { bv = v; best = i; }
            }
            acc += bv;
            s_rate[best] = -3.0e38f;
        }
        const float rob   = acc * (1.0f / (float)CVAR_K);
        const float meanp = s_sp[0] * invN;
        const float brier = s_sb[0] * invN;
        const float cost  = 1.0f + 49.0f * meanp;   // mean(50p + (1-p)) = 1 + 49*mean(p)
        const float cv    = rob - 0.3f;             // CVAR_EPSILON
        const float lam   = lambda_p[0];
        const float lag   = lam * cv;

        out[0] = cost + lag + brier;  // total
        out[1] = cost;                // cost
        out[2] = rob;                 // robustness
        out[3] = brier;               // calibration (brier)
        out[4] = lag;                 // lagrangian term
        out[5] = -lam * cv;           // dual
        out[6] = cv;                  // constraint violation (detached)
    }
}

extern "C" void kernel_launch(void* const* d_in, const int* in_sizes, int n_in,
                              void* d_out, int out_size, void* d_ws, size_t ws_size,
                              hipStream_t stream) {
    const float* p   = (const float*)d_in[0];
    const float* lab = (const float*)d_in[1];
    const float* suc = (const float*)d_in[2];
    const float* lam = (const float*)d_in[3];
    const int*   sd  = (const int*)d_in[4];

    const int n    = in_sizes[0];
    const int nvec = n / 4;             // N = 2^24, divisible by 4
    float* ws = (float*)d_ws;           // needs GRID * PART_STRIDE * 4 = ~540 KB

    rl_partials<<<GRID, BLOCK, 0, stream>>>(p, lab, suc, sd, ws, nvec);
    rl_finalize<<<1, 64, 0, stream>>>(ws, lam, (float*)d_out, GRID, 1.0f / (float)n);
}